// Predict_LSTM_80668075754291
// MI455X (gfx1250) — compile-verified
//
#include <hip/hip_runtime.h>

typedef __attribute__((ext_vector_type(2))) float v2f;
typedef __attribute__((ext_vector_type(8))) float v8f;

#define NB 16384
#define SL 200

struct Params {
  const float* cp;
  const float* Wih0[2]; const float* Whh0[2]; const float* bih0[2]; const float* bhh0[2];
  const float* Wih1[2]; const float* Whh1[2]; const float* bih1[2]; const float* bhh1[2];
  const float* Wlin[2]; const float* blin[2];
  float* out;
};

__device__ __forceinline__ float sigm(float x) {
  return __builtin_amdgcn_rcpf(1.0f + __expf(-x));
}
__device__ __forceinline__ float tanh_f(float x) {
  // tanh(x) = 1 - 2/(exp(2x)+1); stable at both infinities
  return 1.0f - 2.0f * __builtin_amdgcn_rcpf(__expf(2.0f * x) + 1.0f);
}
__device__ __forceinline__ v8f splat8(float x) {
  v8f v = {x, x, x, x, x, x, x, x};
  return v;
}
__device__ __forceinline__ v8f wmma4(v2f a, v2f b, v8f c) {
  return __builtin_amdgcn_wmma_f32_16x16x4_f32(false, a, false, b, (short)0, c, false, false);
}

// Concatenated weight views (row = K index, col picked by lane):
// layer0: K = [x(4) | h0(10) | pad(2)]
__device__ __forceinline__ float w0row(const float* Wih, const float* Whh, int row, int k) {
  if (k < 4)  return Wih[row * 4 + k];
  if (k < 14) return Whh[row * 10 + (k - 4)];
  return 0.0f;
}
// layer1: K = [h0(10) | h1(10)]
__device__ __forceinline__ float w1row(const float* Wih, const float* Whh, int row, int k) {
  if (k < 10) return Wih[row * 10 + k];
  if (k < 20) return Whh[row * 10 + (k - 10)];
  return 0.0f;
}

__launch_bounds__(256, 1)
__global__ void lstm2_wmma(Params p) {
  // per-wave 16x20 f32 transpose tile (row stride 20 -> bank-conflict free)
  __shared__ float smem[8 * 16 * 20];
  const int lane = threadIdx.x & 31;
  const int wave = threadIdx.x >> 5;
  const int lo = lane & 15;        // N (hidden/gate) index, also M for A-loads
  const int hi = lane >> 4;        // half-wave selector
  const int br = blockIdx.y;       // 0 = x branch, 1 = y branch
  const int group = blockIdx.x * 8 + wave;
  const int mbase = group * 16;

  float* lw = smem + wave * 320;

  // zero this wave's LDS slice (initial h0 = h1 = 0)
  for (int i = lane; i < 320; i += 32) lw[i] = 0.0f;
  __builtin_amdgcn_wave_barrier();
  asm volatile("s_wait_dscnt 0x0" ::: "memory");

  // feature gather: branch x -> cols {0,2,4,5}, branch y -> {1,3,4,5}
  // A-quad0 vgpr0 holds k=0 (lanes 0-15) / k=2 (lanes 16-31); vgpr1 k=1 / k=3
  int s0, s1;
  if (br == 0) { s0 = (hi == 0) ? 0 : 4; s1 = (hi == 0) ? 2 : 5; }
  else         { s0 = (hi == 0) ? 1 : 4; s1 = (hi == 0) ? 3 : 5; }
  const int m = mbase + lo;
  const float* xptr0 = p.cp + (size_t)m * (SL * 6) + s0;
  const float* xptr1 = p.cp + (size_t)m * (SL * 6) + s1;

  // Build loop-invariant B operands + bias tiles.
  // B layout (4x16 f32): vgpr0 = row K=base+0 (lanes 0-15) / K=base+2 (16-31),
  //                      vgpr1 = K=base+1 / K=base+3. Lane carries N = lo.
  const float* Wih0 = p.Wih0[br]; const float* Whh0 = p.Whh0[br];
  const float* Wih1 = p.Wih1[br]; const float* Whh1 = p.Whh1[br];
  v2f B0[4][4], B1[4][5];
  float bs0[4], bs1[4];
  {
    const float* bi0 = p.bih0[br]; const float* bh0 = p.bhh0[br];
    const float* bi1 = p.bih1[br]; const float* bh1 = p.bhh1[br];
    const bool ok = lo < 10;
#pragma unroll
    for (int G = 0; G < 4; ++G) {
      const int row = G * 10 + lo;   // PyTorch gate order i,f,g,o
      bs0[G] = ok ? (bi0[G * 10 + lo] + bh0[G * 10 + lo]) : 0.0f;
      bs1[G] = ok ? (bi1[G * 10 + lo] + bh1[G * 10 + lo]) : 0.0f;
#pragma unroll
      for (int q = 0; q < 4; ++q) {
        const int k0 = 4 * q + 2 * hi;
        B0[G][q].x = ok ? w0row(Wih0, Whh0, row, k0)     : 0.0f;
        B0[G][q].y = ok ? w0row(Wih0, Whh0, row, k0 + 1) : 0.0f;
      }
#pragma unroll
      for (int q = 0; q < 5; ++q) {
        const int k0 = 4 * q + 2 * hi;
        B1[G][q].x = ok ? w1row(Wih1, Whh1, row, k0)     : 0.0f;
        B1[G][q].y = ok ? w1row(Wih1, Whh1, row, k0 + 1) : 0.0f;
      }
    }
  }

  v8f c0 = {0, 0, 0, 0, 0, 0, 0, 0};
  v8f c1 = {0, 0, 0, 0, 0, 0, 0, 0};

  // A-load base: element (M=lo, k=col+2*hi); store base: vgpr r -> row r+8*hi, col lo
  float* La = lw + lo * 20 + 2 * hi;
  const int srow = (8 * hi) * 20 + lo;

#pragma unroll 1
  for (int t = 0; t < SL; ++t) {
    // ---- layer 0: A = [x_t | h0_{t-1}] (h pad cols 10,11 hit zero weight rows)
    v2f ax;  ax.x = xptr0[t * 6]; ax.y = xptr1[t * 6];
    v2f ah0 = {La[0], La[1]};
    v2f ah1 = {La[4], La[5]};
    v2f ah2 = {La[8], La[9]};

    v8f g0[4];
#pragma unroll
    for (int G = 0; G < 4; ++G) {
      v8f c = splat8(bs0[G]);
      c = wmma4(ax,  B0[G][0], c);
      c = wmma4(ah0, B0[G][1], c);
      c = wmma4(ah1, B0[G][2], c);
      c = wmma4(ah2, B0[G][3], c);
      g0[G] = c;
    }
    v8f h0;
#pragma unroll
    for (int r = 0; r < 8; ++r) {
      const float iv = sigm(g0[0][r]);
      const float fv = sigm(g0[1][r]);
      const float gv = tanh_f(g0[2][r]);
      const float ov = sigm(g0[3][r]);
      const float cc = fv * c0[r] + iv * gv;
      c0[r] = cc;
      h0[r] = ov * tanh_f(cc);
    }
    // store h0_t into cols 0..9 (predicated so h1_{t-1} at cols 10..19 survives)
    if (lo < 10) {
#pragma unroll
      for (int r = 0; r < 8; ++r) lw[r * 20 + srow] = h0[r];
    }
    __builtin_amdgcn_wave_barrier();
    asm volatile("s_wait_dscnt 0x0" ::: "memory");

    // ---- layer 1: A = [h0_t | h1_{t-1}] (cols 0..19, contiguous K=20)
    v2f a0 = {La[0],  La[1]};
    v2f a1 = {La[4],  La[5]};
    v2f a2 = {La[8],  La[9]};
    v2f a3 = {La[12], La[13]};
    v2f a4 = {La[16], La[17]};

    v8f g1[4];
#pragma unroll
    for (int G = 0; G < 4; ++G) {
      v8f c = splat8(bs1[G]);
      c = wmma4(a0, B1[G][0], c);
      c = wmma4(a1, B1[G][1], c);
      c = wmma4(a2, B1[G][2], c);
      c = wmma4(a3, B1[G][3], c);
      c = wmma4(a4, B1[G][4], c);
      g1[G] = c;
    }
    v8f h1;
#pragma unroll
    for (int r = 0; r < 8; ++r) {
      const float iv = sigm(g1[0][r]);
      const float fv = sigm(g1[1][r]);
      const float gv = tanh_f(g1[2][r]);
      const float ov = sigm(g1[3][r]);
      const float cc = fv * c1[r] + iv * gv;
      c1[r] = cc;
      h1[r] = ov * tanh_f(cc);
    }
    if (lo < 10) {
#pragma unroll
      for (int r = 0; r < 8; ++r) lw[r * 20 + srow + 10] = h1[r];
    }
    __builtin_amdgcn_wave_barrier();
    asm volatile("s_wait_dscnt 0x0" ::: "memory");
  }

  // linear head on final h1 (sitting in LDS cols 10..19)
  if (hi == 0) {
    const float* Wl = p.Wlin[br];
    float acc = p.blin[br][0];
#pragma unroll
    for (int j = 0; j < 10; ++j) acc += lw[lo * 20 + 10 + j] * Wl[j];
    p.out[br * NB + mbase + lo] = acc;
  }
}

extern "C" void kernel_launch(void* const* d_in, const int* in_sizes, int n_in,
                              void* d_out, int out_size, void* d_ws, size_t ws_size,
                              hipStream_t stream) {
  (void)in_sizes; (void)n_in; (void)out_size; (void)d_ws; (void)ws_size;
  Params p;
  p.cp = (const float*)d_in[0];
  for (int b = 0; b < 2; ++b) {
    const int i = 1 + b * 10;
    p.Wih0[b] = (const float*)d_in[i + 0];
    p.Whh0[b] = (const float*)d_in[i + 1];
    p.bih0[b] = (const float*)d_in[i + 2];
    p.bhh0[b] = (const float*)d_in[i + 3];
    p.Wih1[b] = (const float*)d_in[i + 4];
    p.Whh1[b] = (const float*)d_in[i + 5];
    p.bih1[b] = (const float*)d_in[i + 6];
    p.bhh1[b] = (const float*)d_in[i + 7];
    p.Wlin[b] = (const float*)d_in[i + 8];
    p.blin[b] = (const float*)d_in[i + 9];
  }
  p.out = (float*)d_out;
  // 1024 batch-groups per branch, 8 waves/block -> grid (128, 2)
  lstm2_wmma<<<dim3(128, 2), 256, 0, stream>>>(p);
}